// GAT_51762945851517
// MI455X (gfx1250) — compile-verified
//
#include <hip/hip_runtime.h>
#include <hip/hip_bf16.h>
#include <math.h>

// ---------------------------------------------------------------------------
// GAT coordinate network for MI455X (gfx1250, wave32).
// Layers 1/2 of the reference are dead code (h only updated at i==0), so we
// compute: WMMA f32 GEMM (x@W0) -> attention coefs -> 3 edge passes
// (segment-max, segment-sum, weighted aggregate; all L2-resident atomics)
// -> LN/ReLU/row-norm -> per-node MLPs -> center & scale reductions.
// ---------------------------------------------------------------------------

#define D_IN  128
#define HID   64
#define HEADS 8
#define HDIM  8

typedef __attribute__((ext_vector_type(2))) float v2f;
typedef __attribute__((ext_vector_type(8))) float v8f;

// ---------------------------------------------------------------------------
// K1: ht = x @ W0   via V_WMMA_F32_16X16X4_F32.  One wave per 16-row tile,
// 4 col-tiles of 16 (full 64 cols), 32 K-steps of 4.
// A frag (16x4 f32): lane l -> M=l&15, K = k + 2*(l>>4) + {0,1}  (float2 load)
// B frag (4x16 f32): lane l -> N=l&15, K = k + 2*(l>>4) + {0,1}
// C/D (16x16 f32):  vgpr r, lanes 0-15: (M=r, N=l); lanes 16-31: (M=r+8, N=l-16)
// ---------------------------------------------------------------------------
__global__ __launch_bounds__(256) void k_gemm_ht(const float* __restrict__ x,
                                                 const float* __restrict__ W,
                                                 float* __restrict__ ht, int n) {
  const int lane = threadIdx.x & 31;
  const int wave = threadIdx.x >> 5;
  const int tile = blockIdx.x * 8 + wave;
  const int ntiles = (n + 15) >> 4;
  if (tile >= ntiles) return;

  const int m   = lane & 15;
  const int nn  = lane & 15;
  const int kh  = (lane >> 4) << 1;               // 0 or 2
  int row = tile * 16 + m;
  if (row >= n) row = n - 1;                      // safe clamp (n%16==0 anyway)
  const float* xr = x + (size_t)row * D_IN;

  v8f acc0 = {}, acc1 = {}, acc2 = {}, acc3 = {};

  for (int k = 0; k < D_IN; k += 4) {
    if (k + 16 < D_IN) __builtin_prefetch(xr + k + 16, 0, 1);
    v2f a = *(const v2f*)(xr + k + kh);           // 8B-aligned (kh even)
    const float* w0 = W + (size_t)(k + kh) * HID;
    const float* w1 = w0 + HID;
    v2f b0 = { w0[ 0 + nn], w1[ 0 + nn] };
    v2f b1 = { w0[16 + nn], w1[16 + nn] };
    v2f b2 = { w0[32 + nn], w1[32 + nn] };
    v2f b3 = { w0[48 + nn], w1[48 + nn] };
    acc0 = __builtin_amdgcn_wmma_f32_16x16x4_f32(false, a, false, b0, (short)0, acc0, false, false);
    acc1 = __builtin_amdgcn_wmma_f32_16x16x4_f32(false, a, false, b1, (short)0, acc1, false, false);
    acc2 = __builtin_amdgcn_wmma_f32_16x16x4_f32(false, a, false, b2, (short)0, acc2, false, false);
    acc3 = __builtin_amdgcn_wmma_f32_16x16x4_f32(false, a, false, b3, (short)0, acc3, false, false);
  }

  const int rbase = (lane >> 4) << 3;             // 0 or 8
  for (int r = 0; r < 8; ++r) {
    int orow = tile * 16 + rbase + r;
    if (orow >= n) continue;
    float* dst = ht + (size_t)orow * HID + (lane & 15);
    dst[ 0] = acc0[r];
    dst[16] = acc1[r];
    dst[32] = acc2[r];
    dst[48] = acc3[r];
  }
}

// ---------------------------------------------------------------------------
// K2: attention coefficients  al_s[n,h] = <ht[n,h,:], a_src[h,:]>, same for dst
// ---------------------------------------------------------------------------
__global__ void k_alpha_coef(const float* __restrict__ ht,
                             const float* __restrict__ as,
                             const float* __restrict__ ad,
                             float* __restrict__ al_s, float* __restrict__ al_d,
                             int n) {
  int gid = blockIdx.x * blockDim.x + threadIdx.x;      // n*HEADS
  if (gid >= n * HEADS) return;
  int h = gid & 7;
  const float* hp = ht + (size_t)(gid >> 3) * HID + h * HDIM;
  float ss = 0.f, sd = 0.f;
  #pragma unroll
  for (int d = 0; d < HDIM; ++d) {
    ss = fmaf(hp[d], as[h * HDIM + d], ss);
    sd = fmaf(hp[d], ad[h * HDIM + d], sd);
  }
  al_s[gid] = ss;
  al_d[gid] = sd;
}

// ---------------------------------------------------------------------------
// K0: init accumulators
// ---------------------------------------------------------------------------
__global__ void k_init(float* emax, float* den, float* agg, float* red, int n) {
  int i = blockIdx.x * blockDim.x + threadIdx.x;
  if (i < n * HEADS) { emax[i] = -3.0e38f; den[i] = 0.f; }
  if (i < n * HID)   agg[i] = 0.f;
  if (i < 4)         red[i] = 0.f;
}

__device__ __forceinline__ void edge_sd(const int* __restrict__ ei, int E, int e,
                                        int& s, int& d) {
  if (e < E) { s = ei[e]; d = ei[E + e]; }
  else       { s = d = e - E; }                   // appended self-loops
}

__device__ __forceinline__ float lrelu(float v) {
  return v > 0.f ? v : 0.2f * v;
}

// ---------------------------------------------------------------------------
// K3: segment max over dst.  One thread per (edge, head).
// ---------------------------------------------------------------------------
__global__ void k_edge_max(const int* __restrict__ ei,
                           const float* __restrict__ al_s,
                           const float* __restrict__ al_d,
                           float* __restrict__ emax, int E, int total) {
  int gid = blockIdx.x * blockDim.x + threadIdx.x;
  if (gid >= total * HEADS) return;
  int e = gid >> 3, h = gid & 7;
  int s, d; edge_sd(ei, E, e, s, d);
  float ev = lrelu(al_s[s * HEADS + h] + al_d[d * HEADS + h]);
  atomicMax(&emax[d * HEADS + h], ev);
}

// ---------------------------------------------------------------------------
// K4: segment sum of exp(e - emax[dst]).  al/emax arrays are L2-resident;
// recomputing e beats materializing a 105MB per-edge ex array.
// ---------------------------------------------------------------------------
__global__ void k_edge_den(const int* __restrict__ ei,
                           const float* __restrict__ al_s,
                           const float* __restrict__ al_d,
                           const float* __restrict__ emax,
                           float* __restrict__ den, int E, int total) {
  int gid = blockIdx.x * blockDim.x + threadIdx.x;
  if (gid >= total * HEADS) return;
  int e = gid >> 3, h = gid & 7;
  int s, d; edge_sd(ei, E, e, s, d);
  float ev = lrelu(al_s[s * HEADS + h] + al_d[d * HEADS + h]);
  float ex = __expf(ev - emax[d * HEADS + h]);
  atomicAdd(&den[d * HEADS + h], ex);
}

// ---------------------------------------------------------------------------
// K5: weighted aggregate  agg[dst] += alpha * ht[src]
// ---------------------------------------------------------------------------
__global__ void k_edge_agg(const int* __restrict__ ei,
                           const float* __restrict__ al_s,
                           const float* __restrict__ al_d,
                           const float* __restrict__ emax,
                           const float* __restrict__ den,
                           const float* __restrict__ ht,
                           float* __restrict__ agg, int E, int total) {
  int gid = blockIdx.x * blockDim.x + threadIdx.x;
  if (gid >= total * HEADS) return;
  int e = gid >> 3, h = gid & 7;
  int s, d; edge_sd(ei, E, e, s, d);
  float ev = lrelu(al_s[s * HEADS + h] + al_d[d * HEADS + h]);
  float ex = __expf(ev - emax[d * HEADS + h]);
  float alpha = ex / (den[d * HEADS + h] + 1e-16f);
  const float* hs = ht  + (size_t)s * HID + h * HDIM;
  float*       op = agg + (size_t)d * HID + h * HDIM;
  #pragma unroll
  for (int j = 0; j < HDIM; ++j) atomicAdd(&op[j], hs[j] * alpha);
}

__device__ __forceinline__ float wave_sum(float v) {
  #pragma unroll
  for (int m = 16; m >= 1; m >>= 1) v += __shfl_xor(v, m, 32);
  return v;
}
__device__ __forceinline__ float wave_max(float v) {
  #pragma unroll
  for (int m = 16; m >= 1; m >>= 1) v = fmaxf(v, __shfl_xor(v, m, 32));
  return v;
}

// ---------------------------------------------------------------------------
// K6: per-node epilogue: +bias -> LayerNorm -> ReLU -> L2 row-normalize.
// One wave per node, lane handles elements {l, l+32}.
// ---------------------------------------------------------------------------
__global__ __launch_bounds__(256) void k_node_finalize(const float* __restrict__ agg,
                                                       const float* __restrict__ bias,
                                                       const float* __restrict__ g,
                                                       const float* __restrict__ b,
                                                       float* __restrict__ hout, int n) {
  int lane = threadIdx.x & 31;
  int node = (blockIdx.x * blockDim.x + threadIdx.x) >> 5;
  if (node >= n) return;
  const float* ap = agg + (size_t)node * HID;
  float v0 = ap[lane]      + bias[lane];
  float v1 = ap[lane + 32] + bias[lane + 32];
  float mean = wave_sum(v0 + v1) * (1.f / 64.f);
  float d0 = v0 - mean, d1 = v1 - mean;
  float var = wave_sum(d0 * d0 + d1 * d1) * (1.f / 64.f);
  float inv = rsqrtf(var + 1e-5f);
  v0 = fmaxf(d0 * inv * g[lane]      + b[lane],      0.f);
  v1 = fmaxf(d1 * inv * g[lane + 32] + b[lane + 32], 0.f);
  float nrm = sqrtf(wave_sum(v0 * v0 + v1 * v1));
  float invn = 1.f / fmaxf(nrm, 1e-12f);
  hout[(size_t)node * HID + lane]      = v0 * invn;
  hout[(size_t)node * HID + lane + 32] = v1 * invn;
}

// ---------------------------------------------------------------------------
// K7: angle MLP (64->64 LN ReLU ->1) and radius MLP (64->32 LN ReLU ->1,
// sigmoid, r in (0.9,1.1)) -> raw coords.  One wave per node, h row in LDS.
// ---------------------------------------------------------------------------
__global__ __launch_bounds__(256) void k_heads(const float* __restrict__ h,
    const float* __restrict__ aw1, const float* __restrict__ ab1,
    const float* __restrict__ alg, const float* __restrict__ alb,
    const float* __restrict__ aw2, const float* __restrict__ ab2,
    const float* __restrict__ rw1, const float* __restrict__ rb1,
    const float* __restrict__ rlg, const float* __restrict__ rlb,
    const float* __restrict__ rw2, const float* __restrict__ rb2,
    float* __restrict__ coords, int n) {
  __shared__ float sh[8][HID];
  int lane = threadIdx.x & 31;
  int wv   = threadIdx.x >> 5;
  int node = blockIdx.x * 8 + wv;
  bool act = node < n;
  float* hr = sh[wv];
  if (act) {
    hr[lane]      = h[(size_t)node * HID + lane];
    hr[lane + 32] = h[(size_t)node * HID + lane + 32];
  }
  __syncthreads();
  if (!act) return;

  // ---- angle MLP layer 1: z = h @ aw1 + ab1 (lane covers cols l, l+32)
  float z0 = ab1[lane], z1 = ab1[lane + 32];
  for (int k = 0; k < HID; ++k) {
    float hk = hr[k];
    z0 = fmaf(hk, aw1[k * HID + lane],      z0);
    z1 = fmaf(hk, aw1[k * HID + lane + 32], z1);
  }
  float mean = wave_sum(z0 + z1) * (1.f / 64.f);
  float d0 = z0 - mean, d1 = z1 - mean;
  float inv = rsqrtf(wave_sum(d0 * d0 + d1 * d1) * (1.f / 64.f) + 1e-5f);
  z0 = fmaxf(d0 * inv * alg[lane]      + alb[lane],      0.f);
  z1 = fmaxf(d1 * inv * alg[lane + 32] + alb[lane + 32], 0.f);
  float angle = wave_sum(z0 * aw2[lane] + z1 * aw2[lane + 32]) + ab2[0];

  // ---- radius MLP layer 1: 32 cols, lane covers col l
  float r0 = rb1[lane];
  for (int k = 0; k < HID; ++k) r0 = fmaf(hr[k], rw1[k * 32 + lane], r0);
  float rmean = wave_sum(r0) * (1.f / 32.f);
  float rd = r0 - rmean;
  float rinv = rsqrtf(wave_sum(rd * rd) * (1.f / 32.f) + 1e-5f);
  float rz = fmaxf(rd * rinv * rlg[lane] + rlb[lane], 0.f);
  float rr = wave_sum(rz * rw2[lane]) + rb2[0];
  float sig = 1.f / (1.f + __expf(-rr));
  float rad = 0.9f + 0.2f * sig;

  if (lane == 0) {
    coords[(size_t)node * 2 + 0] = rad * __cosf(angle);
    coords[(size_t)node * 2 + 1] = rad * __sinf(angle);
  }
}

// ---------------------------------------------------------------------------
// K8/K9/K10: center then scale by max norm.
// ---------------------------------------------------------------------------
__global__ void k_sum(const float* __restrict__ coords, float* __restrict__ red, int n) {
  int i = blockIdx.x * blockDim.x + threadIdx.x;
  float cx = 0.f, cy = 0.f;
  if (i < n) { cx = coords[2 * i]; cy = coords[2 * i + 1]; }
  cx = wave_sum(cx); cy = wave_sum(cy);
  if ((threadIdx.x & 31) == 0) { atomicAdd(&red[0], cx); atomicAdd(&red[1], cy); }
}

__global__ void k_center_max(float* __restrict__ out, float* __restrict__ red, int n) {
  int i = blockIdx.x * blockDim.x + threadIdx.x;
  float invN = 1.f / (float)n;
  float nr = 0.f;
  if (i < n) {
    float cx = out[2 * i]     - red[0] * invN;
    float cy = out[2 * i + 1] - red[1] * invN;
    out[2 * i]     = cx;
    out[2 * i + 1] = cy;
    nr = sqrtf(cx * cx + cy * cy);
  }
  nr = wave_max(nr);
  if ((threadIdx.x & 31) == 0) atomicMax(&red[2], nr);
}

__global__ void k_scale(float* __restrict__ out, const float* __restrict__ red, int n2) {
  int i = blockIdx.x * blockDim.x + threadIdx.x;
  if (i < n2) out[i] *= 1.f / (red[2] + 1e-8f);
}

// ---------------------------------------------------------------------------
// Host launcher.  d_in order: x, edge_index, then params in dict order:
// [W0,as0,ad0,bias0,lng0,lnb0] x3, aw1,ab1,alg,alb,aw2,ab2, rw1,rb1,rlg,rlb,rw2,rb2
// ---------------------------------------------------------------------------
extern "C" void kernel_launch(void* const* d_in, const int* in_sizes, int n_in,
                              void* d_out, int out_size, void* d_ws, size_t ws_size,
                              hipStream_t stream) {
  const float* x   = (const float*)d_in[0];
  const int*   ei  = (const int*)  d_in[1];
  const float* W0  = (const float*)d_in[2];
  const float* as0 = (const float*)d_in[3];
  const float* ad0 = (const float*)d_in[4];
  const float* bi0 = (const float*)d_in[5];
  const float* lg0 = (const float*)d_in[6];
  const float* lb0 = (const float*)d_in[7];
  // d_in[8..19]: layers 1,2 — dead code in the reference (h never updated)
  const float* aw1 = (const float*)d_in[20];
  const float* ab1 = (const float*)d_in[21];
  const float* alg = (const float*)d_in[22];
  const float* alb = (const float*)d_in[23];
  const float* aw2 = (const float*)d_in[24];
  const float* ab2 = (const float*)d_in[25];
  const float* rw1 = (const float*)d_in[26];
  const float* rb1 = (const float*)d_in[27];
  const float* rlg = (const float*)d_in[28];
  const float* rlb = (const float*)d_in[29];
  const float* rw2 = (const float*)d_in[30];
  const float* rb2 = (const float*)d_in[31];

  const int n = in_sizes[0] / D_IN;      // 100000
  const int E = in_sizes[1] / 2;         // 3200000
  const int total = E + n;               // with self-loops

  // workspace layout (floats)
  float* ws   = (float*)d_ws;
  float* ht   = ws;                      // n*64
  float* als  = ht   + (size_t)n * HID;  // n*8
  float* ald  = als  + (size_t)n * HEADS;
  float* emax = ald  + (size_t)n * HEADS;
  float* den  = emax + (size_t)n * HEADS;
  float* agg  = den  + (size_t)n * HEADS; // n*64
  float* hfin = agg  + (size_t)n * HID;   // n*64
  float* red  = hfin + (size_t)n * HID;   // 4

  float* out = (float*)d_out;

  const int B = 256;
  const int ntiles = (n + 15) / 16;

  k_init<<<(n * HID + B - 1) / B, B, 0, stream>>>(emax, den, agg, red, n);
  k_gemm_ht<<<(ntiles + 7) / 8, B, 0, stream>>>(x, W0, ht, n);
  k_alpha_coef<<<(n * HEADS + B - 1) / B, B, 0, stream>>>(ht, as0, ad0, als, ald, n);

  const int eh = total * HEADS;
  k_edge_max<<<(eh + B - 1) / B, B, 0, stream>>>(ei, als, ald, emax, E, total);
  k_edge_den<<<(eh + B - 1) / B, B, 0, stream>>>(ei, als, ald, emax, den, E, total);
  k_edge_agg<<<(eh + B - 1) / B, B, 0, stream>>>(ei, als, ald, emax, den, ht, agg, E, total);

  k_node_finalize<<<(n * 32 + B - 1) / B, B, 0, stream>>>(agg, bi0, lg0, lb0, hfin, n);
  k_heads<<<(n + 7) / 8, B, 0, stream>>>(hfin, aw1, ab1, alg, alb, aw2, ab2,
                                         rw1, rb1, rlg, rlb, rw2, rb2, out, n);
  k_sum<<<(n + B - 1) / B, B, 0, stream>>>(out, red, n);
  k_center_max<<<(n + B - 1) / B, B, 0, stream>>>(out, red, n);
  k_scale<<<(2 * n + B - 1) / B, B, 0, stream>>>(out, red, 2 * n);
}